// DigitExtractor_62380105007591
// MI455X (gfx1250) — compile-verified
//
#include <hip/hip_runtime.h>
#include <hip/hip_bf16.h>
#include <math.h>

#define TPB   256
#define VEC   4
#define TILE  (TPB * VEC)   // 1024 floats = 4 KB per tile
#define NBUF  8             // async pipeline depth -> 32 KB LDS per block

// ---------------- CDNA5 async global->LDS helpers (ASYNCcnt path) ----------------
// Inline asm per CDNA5 ISA: bypasses the clang builtin whose
// pointer signature differs across toolchains.
//   global_load_async_to_lds_b128  vdst(LDS byte offset), vaddr(64b global), off

__device__ __forceinline__ unsigned lds_offset_of(void* p) {
  return (unsigned)(unsigned long long)
      (__attribute__((address_space(3))) char*)(char*)p;
}

__device__ __forceinline__ void async_ld16(const float4* g, float4* l) {
  unsigned loff = lds_offset_of(l);
  asm volatile("global_load_async_to_lds_b128 %0, %1, off"
               :: "v"(loff), "v"(g)
               : "memory");
}

template <int N>
__device__ __forceinline__ void wait_async_le() {
#if __has_builtin(__builtin_amdgcn_s_wait_asynccnt)
  __builtin_amdgcn_s_wait_asynccnt(N);
#else
  asm volatile("s_wait_asynccnt %0" :: "i"(N) : "memory");
#endif
}

__device__ __forceinline__ void wait_ds0() {
#if __has_builtin(__builtin_amdgcn_s_wait_dscnt)
  __builtin_amdgcn_s_wait_dscnt(0);
#else
  asm volatile("s_wait_dscnt 0" ::: "memory");
#endif
}

// ---------------- math ----------------
// silu_threshold with scale=20:
//   st(y) = (silu(d+10) - silu(d-10)) / 20,  d = 20*y
// Shared-exp form: sig(d+10) = 1/(1 + e^-d * e^-10), sig(d-10) = 1/(1 + e^-d * e^+10)
// -> one v_exp_f32 + two v_rcp_f32 per evaluation.
// Saturation is exact in fp32: y >= ~4.5 -> st == 1.0f bit-exactly, y <= -4.5 -> 0.0f.
__device__ __forceinline__ float st20(float y) {
  float d = 20.0f * y;                // match reference rounding (d computed first)
  float a = d + 10.0f;
  float b = d - 10.0f;
#if __has_builtin(__builtin_amdgcn_exp2f)
  float t = __builtin_amdgcn_exp2f(-1.4426950408889634f * d);   // e^{-d}
#else
  float t = __expf(-d);
#endif
#if __has_builtin(__builtin_amdgcn_rcpf)
  float siga = __builtin_amdgcn_rcpf(__builtin_fmaf(t, 4.5399929762484854e-05f, 1.0f));
  float sigb = __builtin_amdgcn_rcpf(__builtin_fmaf(t, 2.2026465794806718e+04f, 1.0f));
#else
  float siga = 1.0f / __builtin_fmaf(t, 4.5399929762484854e-05f, 1.0f);
  float sigb = 1.0f / __builtin_fmaf(t, 2.2026465794806718e+04f, 1.0f);
#endif
  return (a * siga - b * sigb) * 0.05f;
}

struct DC { float dg, cnt; };

__device__ __forceinline__ DC compute_one(float xv, int n_q, float divisor) {
  const float xa = __builtin_fabsf(xv);

  // ---- count: 1 + sum over thresholds 10..1e5.  For xa > 1e5+4 every term
  // saturates to exactly 1.0f, so count == 6 exactly (predicated skip).
  float cf;
  if (xa < 100004.0f) {
    cf = 1.0f;
    float thr = 10.0f;
#pragma unroll
    for (int i = 0; i < 5; ++i) {
      cf += st20((xa - thr) + 0.5f);
      thr *= 10.0f;
    }
  } else {
    cf = 6.0f;
  }

  // ---- digit: quotient = sum_q L_q * (1 - L_{q+1}) * q  with
  // L_q = st(xa - q*div + 0.5), using upper_q == 1 - L_{q+1} (exact identity).
  // For xa beyond the enumeration window the fp32 result is exactly 0 -> skip
  // (99.9% of uniform [0,1e6] inputs; ~86% of waves skip the whole chain).
  float qf = 0.0f;
  const float limit = (float)n_q * divisor + 4.0f;
  if (xa < limit) {
    float Lq = st20((xa - divisor) + 0.5f);            // L_1
    for (int q = 1; q < n_q; ++q) {
      float Lq1 = st20((xa - (float)(q + 1) * divisor) + 0.5f);
      qf = __builtin_fmaf(Lq * (1.0f - Lq1), (float)q, qf);
      Lq = Lq1;
    }
  }
  float dgf = qf - floorf(qf * 0.1f) * 10.0f;

  DC r;
  r.dg  = (float)(int)dgf;   // truncation, as in astype(int)
  r.cnt = (float)(int)cf;
  return r;
}

// ---------------- kernel ----------------

__global__ __launch_bounds__(TPB)
void DigitExtractor_kernel(const float* __restrict__ x,
                           const int*  __restrict__ posPtr,
                           float* __restrict__ outDigit,
                           float* __restrict__ outCount,
                           int n) {
  __shared__ float4 lbuf[NBUF][TPB];

  // pos is a trace-time scalar in the reference; rebuild divisor / n_q from it.
  int pos = *posPtr;
  pos = pos < 0 ? 0 : (pos > 6 ? 6 : pos);
  int divi = 1;
  for (int i = 0; i < pos; ++i) divi *= 10;
  int n_q = 1000 / divi + 2;
  if (n_q > 1000) n_q = 1000;
  const float divisor = (float)divi;

  const int tid   = threadIdx.x;
  const int b     = blockIdx.x;
  const int G     = gridDim.x;
  const int nFull = n / TILE;
  const int myN   = (b < nFull) ? ((nFull - 1 - b) / G + 1) : 0;
  const float4* xg4 = (const float4*)x;

  if (myN > 0) {
    // Prologue: fill pipeline (no clamped/wasted loads).
    const int P = (myN < NBUF) ? myN : NBUF;
    for (int s = 0; s < P; ++s) {
      async_ld16(&xg4[((size_t)(b) + (size_t)s * G) * TPB + tid], &lbuf[s][tid]);
    }
    for (int k = 0; k < myN; ++k) {
      const int s = k & (NBUF - 1);
      // Wait until the oldest outstanding stage (this one) has landed in LDS.
      const int rem = myN - k;      // tiles left including this one
      if (rem >= NBUF) { wait_async_le<NBUF - 1>(); }
      else {
        switch (rem) {
          case 7:  wait_async_le<6>(); break;
          case 6:  wait_async_le<5>(); break;
          case 5:  wait_async_le<4>(); break;
          case 4:  wait_async_le<3>(); break;
          case 3:  wait_async_le<2>(); break;
          case 2:  wait_async_le<1>(); break;
          default: wait_async_le<0>(); break;
        }
      }
      float4 v = lbuf[s][tid];      // ds_load_b128 (each wave reads its own slice)
      wait_ds0();                   // v in VGPRs -> safe to overwrite the slot
      const int kn = k + NBUF;
      if (kn < myN) {
        async_ld16(&xg4[((size_t)(b) + (size_t)kn * G) * TPB + tid], &lbuf[s][tid]);
      }

      const size_t e0 = (((size_t)(b) + (size_t)k * G) * TPB + tid) * VEC;
      float4 dv, cv;
      DC r0 = compute_one(v.x, n_q, divisor); dv.x = r0.dg; cv.x = r0.cnt;
      DC r1 = compute_one(v.y, n_q, divisor); dv.y = r1.dg; cv.y = r1.cnt;
      DC r2 = compute_one(v.z, n_q, divisor); dv.z = r2.dg; cv.z = r2.cnt;
      DC r3 = compute_one(v.w, n_q, divisor); dv.w = r3.dg; cv.w = r3.cnt;
      *(float4*)(outDigit + e0) = dv;   // global_store_b128
      *(float4*)(outCount + e0) = cv;   // global_store_b128
    }
  }

  // Tail elements (n not a multiple of TILE): plain scalar path.
  for (size_t i = (size_t)nFull * TILE + (size_t)b * TPB + tid;
       i < (size_t)n; i += (size_t)G * TPB) {
    DC r = compute_one(x[i], n_q, divisor);
    outDigit[i] = r.dg;
    outCount[i] = r.cnt;
  }
}

// ---------------- launch ----------------

extern "C" void kernel_launch(void* const* d_in, const int* in_sizes, int n_in,
                              void* d_out, int out_size, void* d_ws, size_t ws_size,
                              hipStream_t stream) {
  const float* x   = (const float*)d_in[0];
  const int*   pos = (const int*)d_in[1];
  const int    n   = in_sizes[0];

  float* out      = (float*)d_out;
  float* outDigit = out;        // reference returns (digit, count)
  float* outCount = out + n;

  const int nFull = n / TILE;
  int grid = nFull / NBUF;      // each block owns >= NBUF tiles so the pipeline fills
  if (grid < 1) grid = 1;
  if (grid > 2048) grid = 2048;

  DigitExtractor_kernel<<<grid, TPB, 0, stream>>>(x, pos, outDigit, outCount, n);

  (void)n_in; (void)out_size; (void)d_ws; (void)ws_size;
}